// LightMetaPolicy_52149492908361
// MI455X (gfx1250) — compile-verified
//
#include <hip/hip_runtime.h>
#include <math.h>

typedef float v2f __attribute__((ext_vector_type(2)));
typedef float v8f __attribute__((ext_vector_type(8)));

// D = A(16x4 f32) * B(4x16 f32) + C(16x16 f32), wave32.
static __device__ __forceinline__ v8f wmma4(v2f a, v2f b, v8f c) {
  return __builtin_amdgcn_wmma_f32_16x16x4_f32(false, a, false, b, (short)0, c,
                                               false, false);
}

static __device__ __forceinline__ float gelu_exact(float x) {
  return 0.5f * x * (1.0f + erff(x * 0.70710678118654752f));
}

// ---------------- workspace float-offset layout ----------------
// [0      .. 10240)  WcatT [64][160]  (k-major; cols: 0-31 Wk, 32-63 Wq, 64-127 Wv, 128-159 Wr@Wq)
// [10240  .. 10400)  bcat  [160]
// [10400  .. 14496)  Wp1T  [64][64]
// [14496  .. 18592)  Wp2T  [64][64]  (cols >=50 zero-padded)
// [18592  .. 22688)  Wh1T  [64][64]
// [22688  .. 22688+B*64)  pooled [B][64]
#define WS_BCAT 10240
#define WS_WP1T 10400
#define WS_WP2T 14496
#define WS_WH1T 18592
#define WS_POOL 22688

__global__ __launch_bounds__(256) void k_prep(
    const float* __restrict__ Wk, const float* __restrict__ bk,
    const float* __restrict__ Wq, const float* __restrict__ bq,
    const float* __restrict__ Wv, const float* __restrict__ bv,
    const float* __restrict__ Wr, const float* __restrict__ br,
    const float* __restrict__ Wp1, const float* __restrict__ Wp2,
    const float* __restrict__ Wh1, float* __restrict__ ws) {
  int idx = blockIdx.x * 256 + threadIdx.x;
  if (idx < 10240) {                       // WcatT[k*160+n]
    int k = idx / 160, n = idx % 160;
    float v;
    if (n < 32)       v = Wk[n * 64 + k];
    else if (n < 64)  v = Wq[(n - 32) * 64 + k];
    else if (n < 128) v = Wv[(n - 64) * 64 + k];
    else {                                 // (Wr @ Wq)[n-128][k]
      int i = n - 128;
      float s = 0.f;
      for (int j = 0; j < 32; ++j) s += Wr[i * 32 + j] * Wq[j * 64 + k];
      v = s;
    }
    ws[idx] = v;
  } else if (idx < WS_WP1T) {              // bcat
    int n = idx - WS_BCAT;
    float v;
    if (n < 32)       v = bk[n];
    else if (n < 64)  v = bq[n - 32];
    else if (n < 128) v = bv[n - 64];
    else {                                 // Wr @ bq + br
      int i = n - 128;
      float s = br[i];
      for (int j = 0; j < 32; ++j) s += Wr[i * 32 + j] * bq[j];
      v = s;
    }
    ws[idx] = v;
  } else if (idx < WS_WP2T) {
    int e = idx - WS_WP1T; int k = e >> 6, n = e & 63;
    ws[idx] = Wp1[n * 64 + k];
  } else if (idx < WS_WH1T) {
    int e = idx - WS_WP2T; int k = e >> 6, n = e & 63;
    ws[idx] = (n < 50) ? Wp2[n * 64 + k] : 0.f;
  } else if (idx < WS_POOL) {
    int e = idx - WS_WH1T; int k = e >> 6, n = e & 63;
    ws[idx] = Wh1[n * 64 + k];
  }
}

// ---------------- front: projection GEMM + attention + pooling ----------------
#define SPB 8            // samples per block
#define ROWS (SPB * 5)   // 40 real agent rows
#define ROWSP 48         // padded to 3 M-tiles of 16

__global__ __launch_bounds__(256) void k_front(const float* __restrict__ x,
                                               const float* __restrict__ ws,
                                               float* __restrict__ pooled) {
  __shared__ float xbuf[ROWSP * 64];     // 12 KB
  __shared__ float pbuf[ROWSP * 160];    // 30 KB
  __shared__ float attnbuf[SPB * 25];
  __shared__ float maskbuf[SPB * 5];

  const int tid = threadIdx.x;
  const int lane = tid & 31;
  const int w = tid >> 5;
  const float* wc = ws;                  // WcatT
  const float* bc = ws + WS_BCAT;

  // stage X tile (coalesced), zero the pad rows
  const long xbase = (long)blockIdx.x * (SPB * 320);
  for (int i = tid; i < SPB * 320; i += 256) xbuf[i] = x[xbase + i];
  for (int i = SPB * 320 + tid; i < ROWSP * 64; i += 256) xbuf[i] = 0.f;
  __syncthreads();

  // GEMM phase: waves 0..2, each a 16-row M-tile, N=160 in 10 tiles, K=64 in 16 steps
  if (w < 3) {
    const int m = lane & 15, h2 = lane >> 4;
    const int row = w * 16 + m;
    v2f a[16];
#pragma unroll
    for (int kk = 0; kk < 16; ++kk) {
      int k0 = 4 * kk + 2 * h2;
      a[kk].x = xbuf[row * 64 + k0];
      a[kk].y = xbuf[row * 64 + k0 + 1];
    }
    for (int nt = 0; nt < 10; ++nt) {
      const int n = nt * 16 + m;
      v2f bfr[16];
#pragma unroll
      for (int kk = 0; kk < 16; ++kk) {
        int k0 = 4 * kk + 2 * h2;
        bfr[kk].x = wc[k0 * 160 + n];
        bfr[kk].y = wc[(k0 + 1) * 160 + n];
      }
      v8f c;
#pragma unroll
      for (int r = 0; r < 8; ++r) c[r] = 0.f;
#pragma unroll
      for (int kk = 0; kk < 16; ++kk) c = wmma4(a[kk], bfr[kk], c);
      const float bb = bc[n];
#pragma unroll
      for (int r = 0; r < 8; ++r)
        pbuf[(w * 16 + r + 8 * h2) * 160 + n] = c[r] + bb;
    }
  }
  __syncthreads();

  // attention phase: wave w owns sample s = w
  const int s = w;
  const int rb = s * 5;
  if (lane < 5) {
    const float* xr = &xbuf[(rb + lane) * 64];
    float ss = 0.f;
    for (int k = 0; k < 64; ++k) ss += fabsf(xr[k]);
    maskbuf[s * 5 + lane] = (ss > 0.01f) ? 1.f : 0.f;
  }
  __syncthreads();

  if (lane < 25) {
    const int i = lane / 5, j = lane % 5;
    const float* pq = &pbuf[(rb + i) * 160];
    const float* pk = &pbuf[(rb + j) * 160];
    float d = 0.f;
    for (int t = 0; t < 32; ++t)
      d += pq[32 + t] * pk[t] + pq[128 + t] * pk[128 + t];
    attnbuf[s * 25 + lane] = d * 0.17677669529663687f;  // 1/sqrt(32)
  }
  __syncthreads();

  float smval = 0.f;
  if (lane < 25) {
    const int i = lane / 5;
    const float* row = &attnbuf[s * 25 + i * 5];
    float mx = row[0];
    for (int j = 1; j < 5; ++j) mx = fmaxf(mx, row[j]);
    float den = 0.f;
    for (int j = 0; j < 5; ++j) den += expf(row[j] - mx);
    smval = expf(attnbuf[s * 25 + lane] - mx) / den;
  }
  __syncthreads();
  if (lane < 25) attnbuf[s * 25 + lane] = smval;
  __syncthreads();

  // context + masked mean pool: each lane handles dims {lane, lane+32}
  float nm = 0.f;
  for (int i = 0; i < 5; ++i) nm += maskbuf[s * 5 + i];
  const float inv = 1.f / (nm + 1e-8f);
  const long gs = (long)blockIdx.x * SPB + s;
#pragma unroll
  for (int half = 0; half < 2; ++half) {
    const int dd = lane + 32 * half;
    float acc = 0.f;
    for (int i = 0; i < 5; ++i) {
      const float mi = maskbuf[s * 5 + i];
      if (mi > 0.f) {
        float ra = 0.f;
        for (int j = 0; j < 5; ++j)
          ra += attnbuf[s * 25 + i * 5 + j] * pbuf[(rb + j) * 160 + 64 + dd];
        acc += ra;
      }
    }
    pooled[gs * 64 + dd] = acc * inv;
  }
}

// ---------------- heads: pooled -> (logits, value) ----------------
__global__ __launch_bounds__(128) void k_heads(
    const float* __restrict__ pooled, const float* __restrict__ wp1T,
    const float* __restrict__ wp2T, const float* __restrict__ wh1T,
    const float* __restrict__ bp1, const float* __restrict__ bp2,
    const float* __restrict__ bh1, const float* __restrict__ Wh2,
    const float* __restrict__ bh2, float* __restrict__ out, int B) {
  __shared__ float ptile[4 * 16 * 64];   // 16 KB
  __shared__ float hbuf[4 * 16 * 64];    // 16 KB

  const int lane = threadIdx.x & 31;
  const int w = threadIdx.x >> 5;
  const int tile = blockIdx.x * 4 + w;   // 16-sample tile per wave
  const long rowBase = (long)tile * 16;
  float* pt = &ptile[w * 1024];
  float* hb = &hbuf[w * 1024];

  // stage pooled tile (coalesced)
  for (int t = 0; t < 32; ++t)
    pt[lane + 32 * t] = pooled[rowBase * 64 + lane + 32 * t];
  __syncthreads();

  const int m = lane & 15, h2 = lane >> 4;

  // A-fragments of pooled tile, reused for both H1 and Hv GEMMs
  v2f a[16];
#pragma unroll
  for (int kk = 0; kk < 16; ++kk) {
    int k0 = 4 * kk + 2 * h2;
    a[kk].x = pt[m * 64 + k0];
    a[kk].y = pt[m * 64 + k0 + 1];
  }

  v8f c1[4], cv[4];
#pragma unroll
  for (int nt = 0; nt < 4; ++nt)
#pragma unroll
    for (int r = 0; r < 8; ++r) { c1[nt][r] = 0.f; cv[nt][r] = 0.f; }

#pragma unroll
  for (int nt = 0; nt < 4; ++nt) {
    const int n = nt * 16 + m;
    v2f b1[16];
#pragma unroll
    for (int kk = 0; kk < 16; ++kk) {
      int k0 = 4 * kk + 2 * h2;
      b1[kk].x = wp1T[k0 * 64 + n];
      b1[kk].y = wp1T[(k0 + 1) * 64 + n];
    }
#pragma unroll
    for (int kk = 0; kk < 16; ++kk) c1[nt] = wmma4(a[kk], b1[kk], c1[nt]);
#pragma unroll
    for (int kk = 0; kk < 16; ++kk) {
      int k0 = 4 * kk + 2 * h2;
      b1[kk].x = wh1T[k0 * 64 + n];
      b1[kk].y = wh1T[(k0 + 1) * 64 + n];
    }
#pragma unroll
    for (int kk = 0; kk < 16; ++kk) cv[nt] = wmma4(a[kk], b1[kk], cv[nt]);
  }

  // bias + exact gelu
#pragma unroll
  for (int nt = 0; nt < 4; ++nt) {
    const int n = nt * 16 + m;
    const float bb1 = bp1[n], bb2 = bh1[n];
#pragma unroll
    for (int r = 0; r < 8; ++r) {
      c1[nt][r] = gelu_exact(c1[nt][r] + bb1);
      cv[nt][r] = gelu_exact(cv[nt][r] + bb2);
    }
  }

  // value head: dot(Hv[s], Wh2) reduced across each half-wave
  float w2n[4];
#pragma unroll
  for (int nt = 0; nt < 4; ++nt) w2n[nt] = Wh2[nt * 16 + m];
  float vacc[8];
#pragma unroll
  for (int r = 0; r < 8; ++r)
    vacc[r] = cv[0][r] * w2n[0] + cv[1][r] * w2n[1] + cv[2][r] * w2n[2] +
              cv[3][r] * w2n[3];
  for (int off = 1; off < 16; off <<= 1)
#pragma unroll
    for (int r = 0; r < 8; ++r) vacc[r] += __shfl_xor(vacc[r], off, 32);
  if (m == 0) {
    const float bv0 = bh2[0];
    const long vOff = (long)B * 50;
#pragma unroll
    for (int r = 0; r < 8; ++r)
      out[vOff + rowBase + r + 8 * h2] = vacc[r] + bv0;
  }

  // H1 -> LDS (D layout -> row major) for the logits GEMM
#pragma unroll
  for (int nt = 0; nt < 4; ++nt)
#pragma unroll
    for (int r = 0; r < 8; ++r)
      hb[(r + 8 * h2) * 64 + nt * 16 + m] = c1[nt][r];
  __syncthreads();

  v2f a2[16];
#pragma unroll
  for (int kk = 0; kk < 16; ++kk) {
    int k0 = 4 * kk + 2 * h2;
    a2[kk].x = hb[m * 64 + k0];
    a2[kk].y = hb[m * 64 + k0 + 1];
  }
  v8f cl[4];
#pragma unroll
  for (int nt = 0; nt < 4; ++nt)
#pragma unroll
    for (int r = 0; r < 8; ++r) cl[nt][r] = 0.f;
#pragma unroll
  for (int nt = 0; nt < 4; ++nt) {
    const int n = nt * 16 + m;
    v2f b2[16];
#pragma unroll
    for (int kk = 0; kk < 16; ++kk) {
      int k0 = 4 * kk + 2 * h2;
      b2[kk].x = wp2T[k0 * 64 + n];
      b2[kk].y = wp2T[(k0 + 1) * 64 + n];
    }
#pragma unroll
    for (int kk = 0; kk < 16; ++kk) cl[nt] = wmma4(a2[kk], b2[kk], cl[nt]);
  }
  // store logits (only the 50 valid columns)
#pragma unroll
  for (int nt = 0; nt < 4; ++nt) {
    const int n = nt * 16 + m;
    if (n < 50) {
      const float bb = bp2[n];
#pragma unroll
      for (int r = 0; r < 8; ++r)
        out[(rowBase + r + 8 * h2) * 50 + n] = cl[nt][r] + bb;
    }
  }
}

extern "C" void kernel_launch(void* const* d_in, const int* in_sizes, int n_in,
                              void* d_out, int out_size, void* d_ws,
                              size_t ws_size, hipStream_t stream) {
  const float* x   = (const float*)d_in[0];
  const float* Wk  = (const float*)d_in[1];
  const float* bk  = (const float*)d_in[2];
  const float* Wq  = (const float*)d_in[3];
  const float* bq  = (const float*)d_in[4];
  const float* Wv  = (const float*)d_in[5];
  const float* bv  = (const float*)d_in[6];
  const float* Wr  = (const float*)d_in[7];
  const float* br  = (const float*)d_in[8];
  const float* Wp1 = (const float*)d_in[9];
  const float* bp1 = (const float*)d_in[10];
  const float* Wp2 = (const float*)d_in[11];
  const float* bp2 = (const float*)d_in[12];
  const float* Wh1 = (const float*)d_in[13];
  const float* bh1 = (const float*)d_in[14];
  const float* Wh2 = (const float*)d_in[15];
  const float* bh2 = (const float*)d_in[16];
  float* ws = (float*)d_ws;
  float* out = (float*)d_out;
  const int B = in_sizes[0] / 320;

  k_prep<<<(WS_POOL + 255) / 256, 256, 0, stream>>>(Wk, bk, Wq, bq, Wv, bv, Wr,
                                                    br, Wp1, Wp2, Wh1, ws);
  float* pooled = ws + WS_POOL;
  k_front<<<B / SPB, 256, 0, stream>>>(x, ws, pooled);
  k_heads<<<B / 64, 128, 0, stream>>>(pooled, ws + WS_WP1T, ws + WS_WP2T,
                                      ws + WS_WH1T, bp1, bp2, bh1, Wh2, bh2,
                                      out, B);
}